// RMSDLoss_84138409328676
// MI455X (gfx1250) — compile-verified
//
#include <hip/hip_runtime.h>
#include <hip/hip_bf16.h>

typedef __attribute__((ext_vector_type(2))) float v2f;
typedef __attribute__((ext_vector_type(8))) float v8f;

#define N_RES   1024
#define N_ATOMS 37
#define EPS_F   1e-8f
#define WAVES   8                          // waves per block (one block per batch)
#define CHUNKS  (N_RES / (4 * WAVES))      // 32 K=4 chunks per wave (compile-time)

// One block (8 waves) per batch. Each wave accumulates a partial Gram
// G_w = M_w^T M_w over its residue subset with V_WMMA_F32_16X16X4_F32
// (K=4 residues per WMMA, 32 WMMAs per wave), the 8 partial 16x16 tiles are
// summed in LDS, then thread 0 does the closed-form 3x3 Kabsch RMSD.
//
// M columns per residue: [p_x p_y p_z t_x t_y t_z mask] (rows 7..15 zero).
// G = M^T M then contains Sum(p_i t_j), Sum|p|^2, Sum|t|^2, Sum p, Sum t, N.
//
// ISA layouts (CDNA5 ISA):
//   A 16x4 f32 : lane L row M=L%16; VGPR0 = K=(L<16?0:2), VGPR1 = K=(L<16?1:3)
//   B 4x16 f32 : VGPR0 = row K=(L<16?0:2) col N=L%16, VGPR1 = K=(L<16?1:3)
//   => per-lane A and B elements are IDENTICAL for a Gram product.
//   C/D 16x16 f32: VGPR v, lane L -> (M = v + 8*(L>=16), N = L%16)
__global__ __launch_bounds__(32 * WAVES) void kabsch_gram_kernel(
    const float* __restrict__ pred,
    const float* __restrict__ truec,
    const float* __restrict__ amask,
    float* __restrict__ rmsd_out)
{
    const int b    = blockIdx.x;
    const int tid  = threadIdx.x;
    const int wave = tid >> 5;      // 0..7
    const int lane = tid & 31;      // 0..31
    const int r    = lane & 15;     // row of M^T (0..2 p, 3..5 t, 6 mask)
    const int hi   = lane >> 4;

    const size_t coordBase = (size_t)b * N_RES * N_ATOMS * 3 + 3; // atom CA=1
    const size_t maskBase  = (size_t)b * N_RES * N_ATOMS + 1;

    // Per-lane source select (re-converges before the WMMA loop, so EXEC is
    // all-ones for every v_wmma).
    const float scale = (r < 7) ? 1.0f : 0.0f;
    const int   rr    = (r < 7) ? r : 6;
    const float* ptr;
    int stride;
    if (rr < 3)      { ptr = pred  + coordBase + rr;       stride = N_ATOMS * 3; }
    else if (rr < 6) { ptr = truec + coordBase + (rr - 3); stride = N_ATOMS * 3; }
    else             { ptr = amask + maskBase;             stride = N_ATOMS;     }

    // Fold the per-lane K-half offset and this wave's first chunk into the
    // base pointer; then the loop is a pure pointer-increment stream with a
    // compile-time trip count (no exec-masked loop, no 64-bit multiplies).
    ptr += (size_t)(hi * 2 + wave * 4) * stride;
    const size_t step = (size_t)(4 * WAVES) * stride;   // residues per block pass

    v8f acc = {};
#pragma unroll 4
    for (int i = 0; i < CHUNKS; ++i) {
        const float v0 = ptr[0] * scale;                // K = chunk + 2*hi
        const float v1 = ptr[(size_t)stride] * scale;   // K = chunk + 2*hi + 1
        v2f ab; ab.x = v0; ab.y = v1;
        acc = __builtin_amdgcn_wmma_f32_16x16x4_f32(
            /*neg_a=*/false, ab, /*neg_b=*/false, ab,
            /*c_mod=*/(short)0, acc, /*reuse_a=*/false, /*reuse_b=*/false);
        ptr += step;
    }

    // Spill each wave's 16x16 tile to LDS and sum the 8 partials.
    __shared__ float Gp[WAVES][16 * 16];
    __shared__ float G[16 * 16];
#pragma unroll
    for (int v = 0; v < 8; ++v) {
        const int row = v + hi * 8;
        Gp[wave][row * 16 + r] = acc[v];
    }
    __syncthreads();
    {
        // 256 threads, one Gram entry each: sum over 8 waves.
        float s = 0.0f;
#pragma unroll
        for (int w = 0; w < WAVES; ++w) s += Gp[w][tid];
        G[tid] = s;
    }
    __syncthreads();

    if (tid == 0) {
        const float Mn    = G[6 * 16 + 6];          // sum of mask
        const float denom = Mn + EPS_F;
        float Sp[3], St[3], cx[3], dx[3];
#pragma unroll
        for (int i = 0; i < 3; ++i) {
            Sp[i] = G[i * 16 + 6];
            St[i] = G[(3 + i) * 16 + 6];
            cx[i] = Sp[i] / denom;                  // pred centroid
            dx[i] = St[i] / denom;                  // true centroid
        }
        const float Spp = G[0] + G[17] + G[34];                       // sum|p|^2
        const float Stt = G[3*16+3] + G[4*16+4] + G[5*16+5];          // sum|t|^2

        // Centered covariance H_ij = sum (p_i - c_i)(t_j - d_j)
        float H[3][3];
#pragma unroll
        for (int i = 0; i < 3; ++i)
#pragma unroll
            for (int j = 0; j < 3; ++j)
                H[i][j] = G[i * 16 + 3 + j] - cx[i] * St[j] - dx[j] * Sp[i]
                        + Mn * cx[i] * dx[j];

        const float Ep = Spp - 2.0f*(cx[0]*Sp[0]+cx[1]*Sp[1]+cx[2]*Sp[2])
                       + Mn*(cx[0]*cx[0]+cx[1]*cx[1]+cx[2]*cx[2]);
        const float Et = Stt - 2.0f*(dx[0]*St[0]+dx[1]*St[1]+dx[2]*St[2])
                       + Mn*(dx[0]*dx[0]+dx[1]*dx[1]+dx[2]*dx[2]);

        const float detH =
              H[0][0]*(H[1][1]*H[2][2] - H[1][2]*H[2][1])
            - H[0][1]*(H[1][0]*H[2][2] - H[1][2]*H[2][0])
            + H[0][2]*(H[1][0]*H[2][1] - H[1][1]*H[2][0]);

        // K = H^T H (symmetric PSD); eigenvalues in closed form.
        float K[3][3];
#pragma unroll
        for (int i = 0; i < 3; ++i)
#pragma unroll
            for (int j = 0; j < 3; ++j)
                K[i][j] = H[0][i]*H[0][j] + H[1][i]*H[1][j] + H[2][i]*H[2][j];

        const float p1 = K[0][1]*K[0][1] + K[0][2]*K[0][2] + K[1][2]*K[1][2];
        const float q  = (K[0][0] + K[1][1] + K[2][2]) * (1.0f / 3.0f);
        const float p2 = (K[0][0]-q)*(K[0][0]-q) + (K[1][1]-q)*(K[1][1]-q)
                       + (K[2][2]-q)*(K[2][2]-q) + 2.0f * p1;
        const float p  = sqrtf(fmaxf(p2, 0.0f) * (1.0f / 6.0f) + 1e-30f);
        const float ip = 1.0f / p;
        // Bm = (K - q I) / p ; rdet = det(Bm)/2, clamped into acos domain
        float Bm[3][3];
#pragma unroll
        for (int i = 0; i < 3; ++i)
#pragma unroll
            for (int j = 0; j < 3; ++j)
                Bm[i][j] = (K[i][j] - ((i == j) ? q : 0.0f)) * ip;
        float rdet = 0.5f * (
              Bm[0][0]*(Bm[1][1]*Bm[2][2] - Bm[1][2]*Bm[2][1])
            - Bm[0][1]*(Bm[1][0]*Bm[2][2] - Bm[1][2]*Bm[2][0])
            + Bm[0][2]*(Bm[1][0]*Bm[2][1] - Bm[1][1]*Bm[2][0]));
        rdet = fminf(1.0f, fmaxf(-1.0f, rdet));
        const float phi = acosf(rdet) * (1.0f / 3.0f);
        const float e1 = q + 2.0f * p * cosf(phi);                    // largest
        const float e3 = q + 2.0f * p * cosf(phi + 2.0943951024f);    // smallest
        const float e2 = 3.0f * q - e1 - e3;

        const float s1 = sqrtf(fmaxf(e1, 0.0f));
        const float s2 = sqrtf(fmaxf(e2, 0.0f));
        const float s3 = sqrtf(fmaxf(e3, 0.0f));
        const float sgn = (detH < 0.0f) ? -1.0f : 1.0f;
        const float trRH = s1 + s2 + sgn * s3;

        const float sumsq = fmaxf(Ep + Et - 2.0f * trRH, 0.0f);
        rmsd_out[b] = sqrtf(sumsq / denom);
    }
}

// Deterministic mean over B per-batch RMSDs (no atomics; d_out fully rewritten).
__global__ __launch_bounds__(256) void mean_kernel(
    const float* __restrict__ rmsd, float* __restrict__ out, int B)
{
    __shared__ float s[256];
    const int t = threadIdx.x;
    float v = 0.0f;
    for (int i = t; i < B; i += 256) v += rmsd[i];
    s[t] = v;
    __syncthreads();
    for (int off = 128; off > 0; off >>= 1) {
        if (t < off) s[t] += s[t + off];
        __syncthreads();
    }
    if (t == 0) out[0] = s[0] / (float)B;
}

extern "C" void kernel_launch(void* const* d_in, const int* in_sizes, int n_in,
                              void* d_out, int out_size, void* d_ws, size_t ws_size,
                              hipStream_t stream) {
    const float* pred  = (const float*)d_in[0];
    const float* truec = (const float*)d_in[1];
    const float* amask = (const float*)d_in[2];
    const int B = in_sizes[2] / (N_RES * N_ATOMS);   // 256

    float* rmsd_ws = (float*)d_ws;                    // B floats of scratch
    kabsch_gram_kernel<<<B, 32 * WAVES, 0, stream>>>(pred, truec, amask, rmsd_ws);
    mean_kernel<<<1, 256, 0, stream>>>(rmsd_ws, (float*)d_out, B);
}